// BatchHardTripletLoss_35914516529588
// MI455X (gfx1250) — compile-verified
//
#include <hip/hip_runtime.h>
#include <math.h>

#define D 128
#define MARGIN_F 0.2f
#define EPS_F 1e-6f
#define BIGF 3.4e38f

typedef __attribute__((ext_vector_type(2))) float v2f;
typedef __attribute__((ext_vector_type(8))) float v8f;

// ---------------- Kernel 0: row squared norms + zero the loss accumulator ----------------
__global__ void prep_kernel(const float* __restrict__ f, float* __restrict__ sq,
                            float* __restrict__ accum, int n) {
    int i = blockIdx.x * blockDim.x + threadIdx.x;
    if (i == 0) accum[0] = 0.0f;
    if (i < n) {
        const float4* p = (const float4*)(f + (size_t)i * D);
        float s = 0.0f;
#pragma unroll
        for (int k = 0; k < D / 4; ++k) {
            float4 x = p[k];
            s += x.x * x.x + x.y * x.y + x.z * x.z + x.w * x.w;
        }
        sq[i] = s;
    }
}

// ---------------- Kernel 1: fused WMMA Gram matrix + batch-hard mining ----------------
// One block = 16 rows (one M-tile). 8 waves split the 512 N-tiles; each wave runs two
// independent WMMA accumulator chains (K=128 -> 32 chained v_wmma_f32_16x16x4_f32 each).
// All 64 B-fragment loads of a tile-pair are forced into registers BEFORE the WMMA
// chain so they issue as load clauses and overlap the compute (instead of the
// load->wait->wmma lockstep the scheduler produced otherwise). Distances and
// argmax/argmin mining are fused in the epilogue: the 256MB distance matrix never
// touches memory.
__global__ __launch_bounds__(256) void mine_kernel(const float* __restrict__ F,
                                                   const int* __restrict__ ident,
                                                   const float* __restrict__ sq,
                                                   int* __restrict__ posi,
                                                   int* __restrict__ negi, int n) {
    const int lane  = threadIdx.x & 31;
    const int wv    = threadIdx.x >> 5;   // wave in block, 0..7
    const int hi    = lane >> 4;          // half-wave: 0/1
    const int nl    = lane & 15;
    const int rbase = blockIdx.x * 16;

    // A fragments for the 16-row strip, WMMA 16x16x4 A layout:
    // lane L holds row M = L%16, VGPR v holds K = 4k + 2*(L/16) + v.
    v2f afrag[32];
    const v2f* arow = (const v2f*)(F + (size_t)(rbase + nl) * D + 2 * hi);
#pragma unroll
    for (int k = 0; k < 32; ++k) afrag[k] = arow[2 * k];

    // Per-lane row metadata for the C/D layout: row = rbase + v + 8*hi.
    float sqi[8]; int idi[8];
#pragma unroll
    for (int v = 0; v < 8; ++v) {
        int r = rbase + v + 8 * hi;
        sqi[v] = sq[r];
        idi[v] = ident[r];
    }

    float bp[8], bn[8]; int bpj[8], bnj[8];
#pragma unroll
    for (int v = 0; v < 8; ++v) { bp[v] = -1.0f; bn[v] = BIGF; bpj[v] = 0; bnj[v] = 0; }

    const int ntiles = n >> 4;  // 512 column tiles

    auto epilogue = [&](const v8f& acc, int jt) {
        int jc = jt * 16 + nl;     // column index this lane owns
        float sqj = sq[jc];
        int idj = ident[jc];
#pragma unroll
        for (int v = 0; v < 8; ++v) {
            float d2 = sqi[v] + sqj - 2.0f * acc[v];
            float d = sqrtf(fmaxf(d2, 0.0f));
            bool same = (idi[v] == idj);
            float pvv = same ? d : 0.0f;
            if (pvv > bp[v]) { bp[v] = pvv; bpj[v] = jc; }   // strict > keeps first idx
            float nvv = same ? BIGF : d;
            if (nvv < bn[v]) { bn[v] = nvv; bnj[v] = jc; }
        }
    };

    // Wave wv covers tiles { wv, wv+8, wv+16, ... } in increasing order (monotone j per
    // lane -> first-index tiebreak preserved), two chains per iteration.
    const int nit = ntiles / 16;
    for (int it = 0; it < nit; ++it) {
        int jt0 = wv + it * 16;
        int jt1 = jt0 + 8;
        const v2f* brow0 = (const v2f*)(F + (size_t)(jt0 * 16 + nl) * D + 2 * hi);
        const v2f* brow1 = (const v2f*)(F + (size_t)(jt1 * 16 + nl) * D + 2 * hi);

        // Preload both 16x128 B tiles into registers: 64 global_load_b64 issued as
        // clauses up front; the WMMA chain below drains them with decreasing
        // s_wait_loadcnt thresholds, overlapping L2 latency with matrix math.
        v2f b0[32], b1[32];
#pragma unroll
        for (int k = 0; k < 32; ++k) {
            b0[k] = brow0[2 * k];
            b1[k] = brow1[2 * k];
        }

        // Warm the WGP$ for the next iteration's B rows (global_prefetch_b8).
        if (it + 1 < nit) {
            const char* p0 = (const char*)(F + (size_t)((jt0 + 16) * 16 + nl) * D + 2 * hi);
            const char* p1 = (const char*)(F + (size_t)((jt1 + 16) * 16 + nl) * D + 2 * hi);
            __builtin_prefetch(p0, 0, 0);
            __builtin_prefetch(p0 + 256, 0, 0);
            __builtin_prefetch(p1, 0, 0);
            __builtin_prefetch(p1 + 256, 0, 0);
        }

        v8f acc0 = {0.f, 0.f, 0.f, 0.f, 0.f, 0.f, 0.f, 0.f};
        v8f acc1 = {0.f, 0.f, 0.f, 0.f, 0.f, 0.f, 0.f, 0.f};
#pragma unroll
        for (int k = 0; k < 32; ++k) {
            acc0 = __builtin_amdgcn_wmma_f32_16x16x4_f32(false, afrag[k], false, b0[k],
                                                         (short)0, acc0, false, false);
            acc1 = __builtin_amdgcn_wmma_f32_16x16x4_f32(false, afrag[k], false, b1[k],
                                                         (short)0, acc1, false, false);
        }
        epilogue(acc0, jt0);
        epilogue(acc1, jt1);
    }

    // Cross-lane reduce: each half-wave's 16 lanes hold the 16 N-columns of one row.
    __shared__ float mpv[16][8]; __shared__ int mpj[16][8];
    __shared__ float mnv[16][8]; __shared__ int mnj[16][8];
#pragma unroll
    for (int v = 0; v < 8; ++v) {
        float pv = bp[v]; int pj = bpj[v];
        float nv = bn[v]; int nj = bnj[v];
#pragma unroll
        for (int m = 1; m < 16; m <<= 1) {
            float opv = __shfl_xor(pv, m, 16);
            int   opj = __shfl_xor(pj, m, 16);
            if (opv > pv || (opv == pv && opj < pj)) { pv = opv; pj = opj; }
            float onv = __shfl_xor(nv, m, 16);
            int   onj = __shfl_xor(nj, m, 16);
            if (onv < nv || (onv == nv && onj < nj)) { nv = onv; nj = onj; }
        }
        if (nl == 0) {
            int row = v + 8 * hi;
            mpv[row][wv] = pv; mpj[row][wv] = pj;
            mnv[row][wv] = nv; mnj[row][wv] = nj;
        }
    }
    __syncthreads();

    // Merge the 8 waves' partial results (j ranges interleave -> tiebreak on index).
    if (threadIdx.x < 16) {
        int row = threadIdx.x;
        float pv = mpv[row][0]; int pj = mpj[row][0];
        float nv = mnv[row][0]; int nj = mnj[row][0];
#pragma unroll
        for (int w2 = 1; w2 < 8; ++w2) {
            float opv = mpv[row][w2]; int opj = mpj[row][w2];
            if (opv > pv || (opv == pv && opj < pj)) { pv = opv; pj = opj; }
            float onv = mnv[row][w2]; int onj = mnj[row][w2];
            if (onv < nv || (onv == nv && onj < nj)) { nv = onv; nj = onj; }
        }
        posi[rbase + row] = pj;
        negi[rbase + row] = nj;
    }
}

// ---------------- Kernel 2: pairwise distance (with eps) + hinge, atomic reduce ----------------
__global__ __launch_bounds__(256) void loss_kernel(const float* __restrict__ F,
                                                   const int* __restrict__ posi,
                                                   const int* __restrict__ negi,
                                                   float* __restrict__ accum, int n) {
    int gw = (blockIdx.x * blockDim.x + threadIdx.x) >> 5;  // one wave per row
    int lane = threadIdx.x & 31;
    if (gw >= n) return;
    int pi = posi[gw], ni = negi[gw];
    const float4* a = (const float4*)(F + (size_t)gw * D);
    const float4* p = (const float4*)(F + (size_t)pi * D);
    const float4* q = (const float4*)(F + (size_t)ni * D);
    float4 av = a[lane], pv = p[lane], qv = q[lane];
    float dx, sp = 0.0f, sn = 0.0f;
    dx = av.x - pv.x + EPS_F; sp += dx * dx;
    dx = av.y - pv.y + EPS_F; sp += dx * dx;
    dx = av.z - pv.z + EPS_F; sp += dx * dx;
    dx = av.w - pv.w + EPS_F; sp += dx * dx;
    dx = av.x - qv.x + EPS_F; sn += dx * dx;
    dx = av.y - qv.y + EPS_F; sn += dx * dx;
    dx = av.z - qv.z + EPS_F; sn += dx * dx;
    dx = av.w - qv.w + EPS_F; sn += dx * dx;
#pragma unroll
    for (int m = 1; m < 32; m <<= 1) {
        sp += __shfl_xor(sp, m, 32);
        sn += __shfl_xor(sn, m, 32);
    }
    if (lane == 0) {
        float l = fmaxf(MARGIN_F + sqrtf(sp) - sqrtf(sn), 0.0f);
        atomicAdd(accum, l);
    }
}

// ---------------- Kernel 3: mean ----------------
__global__ void final_kernel(const float* __restrict__ accum, float* __restrict__ out, int n) {
    out[0] = accum[0] / (float)n;
}

extern "C" void kernel_launch(void* const* d_in, const int* in_sizes, int n_in,
                              void* d_out, int out_size, void* d_ws, size_t ws_size,
                              hipStream_t stream) {
    const float* feature = (const float*)d_in[0];
    const int*   ident   = (const int*)d_in[1];
    int n = in_sizes[1];  // 8192

    float* sq    = (float*)d_ws;
    int*   posi  = (int*)(sq + n);
    int*   negi  = posi + n;
    float* accum = (float*)(negi + n);
    float* out   = (float*)d_out;

    prep_kernel<<<(n + 255) / 256, 256, 0, stream>>>(feature, sq, accum, n);
    mine_kernel<<<n / 16, 256, 0, stream>>>(feature, ident, sq, posi, negi, n);
    loss_kernel<<<(n * 32 + 255) / 256, 256, 0, stream>>>(feature, posi, negi, accum, n);
    final_kernel<<<1, 1, 0, stream>>>(accum, out, n);
}